// IncepGCN_60284160966677
// MI455X (gfx1250) — compile-verified
//
#include <hip/hip_runtime.h>
#include <hip/hip_bf16.h>

typedef __attribute__((ext_vector_type(2))) float v2f;
typedef __attribute__((ext_vector_type(8))) float v8f;

static __host__ int cdiv(long a, long b) { return (int)((a + b - 1) / b); }

// ---------------------------------------------------------------------------
// small elementwise kernels
// ---------------------------------------------------------------------------
__global__ void k_fill(float* __restrict__ p, float v, long n) {
    long i = (long)blockIdx.x * blockDim.x + threadIdx.x;
    if (i < n) p[i] = v;
}

__global__ void k_deg_accum(const int* __restrict__ dst, const float* __restrict__ ew,
                            float* __restrict__ deg, int E) {
    int e = blockIdx.x * blockDim.x + threadIdx.x;
    if (e < E) unsafeAtomicAdd(&deg[dst[e]], ew[e]);
}

__global__ void k_dinv(float* __restrict__ deg, int N) {
    int i = blockIdx.x * blockDim.x + threadIdx.x;
    if (i < N) {
        float d = deg[i];
        deg[i] = (d > 0.0f) ? rsqrtf(d) : 0.0f;   // deg buffer becomes dinv
    }
}

__global__ void k_norm(const int* __restrict__ src, const int* __restrict__ dst,
                       const float* __restrict__ ew, const float* __restrict__ dinv,
                       float* __restrict__ normv, int E) {
    int e = blockIdx.x * blockDim.x + threadIdx.x;
    if (e < E) normv[e] = dinv[src[e]] * ew[e] * dinv[dst[e]];
}

// out[i,:] = b[:] + dinv[i]^2 * hw[i,:]   (self-loop term + bias, pre-scatter init)
__global__ void k_self_init(const float* __restrict__ hw, const float* __restrict__ dinv,
                            const float* __restrict__ b, float* __restrict__ out,
                            int N, int H) {
    int hq = H >> 2;                                   // float4s per row (32 for H=128)
    long gid = (long)blockIdx.x * blockDim.x + threadIdx.x;
    long i = gid / hq;
    int  q = (int)(gid % hq);
    if (i >= N) return;
    float di = dinv[i];
    float s = di * di;
    float4 v  = ((const float4*)(hw + i * (long)H))[q];
    float4 bv = ((const float4*)b)[q];
    float4 o;
    o.x = bv.x + s * v.x; o.y = bv.y + s * v.y;
    o.z = bv.z + s * v.z; o.w = bv.w + s * v.w;
    ((float4*)(out + i * (long)H))[q] = o;
}

// one wave per edge: out[dst,:] += norm[e] * hw[src,:]   (L2-resident fp32 atomics)
__global__ void k_edge_agg(const int* __restrict__ src, const int* __restrict__ dst,
                           const float* __restrict__ normv, const float* __restrict__ hw,
                           float* __restrict__ out, int E, int H) {
    int w    = (int)(((long)blockIdx.x * blockDim.x + threadIdx.x) >> 5);
    int lane = threadIdx.x & 31;
    if (w >= E) return;
    int   s  = src[w];
    int   d  = dst[w];
    float nv = normv[w];
    const float4* hrow = (const float4*)(hw + (long)s * H);
    float*        orow = out + (long)d * H;
    for (int q = lane; q < (H >> 2); q += 32) {
        float4 v = hrow[q];
        float* o = orow + q * 4;
        unsafeAtomicAdd(o + 0, nv * v.x);
        unsafeAtomicAdd(o + 1, nv * v.y);
        unsafeAtomicAdd(o + 2, nv * v.z);
        unsafeAtomicAdd(o + 3, nv * v.w);
    }
}

__global__ void k_init_out(float* __restrict__ out, const float* __restrict__ bias,
                           long N, int C) {
    long gid = (long)blockIdx.x * blockDim.x + threadIdx.x;
    if (gid < N * (long)C) out[gid] = bias[gid % C];
}

// ---------------------------------------------------------------------------
// fp32 WMMA GEMM:  Cout(M x Ncols) {=, +=} A(M x K, lda) @ B(K x Ncols, ldb) [+ bias]
//   grid.x = ceil(M / (16*MT)) row blocks; one wave per 16-column tile.
//   4 m-tiles per block: each B fragment feeds 4 WMMAs (4x B reuse).
//   A staged in LDS (row stride 132 -> conflict-free column reads, ds_load_b64).
//   B loads are branch-free: column index clamped, invalid columns compute
//   garbage into their own lanes and are masked at store (columns independent).
//   K must be a multiple of 128 (512 / 128 here).
// ---------------------------------------------------------------------------
#define MT 4
#define GEMM_KC 128
#define LDS_STRIDE 132

__global__ void k_gemm_f32_wmma(const float* __restrict__ A, int lda,
                                const float* __restrict__ B, int ldb,
                                const float* __restrict__ bias,
                                float* __restrict__ Cout, int ldc,
                                int M, int K, int Ncols, int accumulate) {
    __shared__ float As[16 * MT * LDS_STRIDE];

    const int m0     = blockIdx.x * (16 * MT);
    const int tid    = threadIdx.x;
    const int lane   = tid & 31;
    const int wave   = tid >> 5;
    const int ntiles = (Ncols + 15) >> 4;

    const int mr    = lane & 15;            // row within m-tile / col within n-tile
    const int khalf = (lane >> 4) << 1;     // 0 for lanes 0-15, 2 for lanes 16-31

    const int  col   = wave * 16 + mr;
    const bool colok = (wave < ntiles) && (col < Ncols);
    const int  col_ld = (col < Ncols) ? col : (Ncols - 1);   // clamped load column

    v8f acc[MT];
    #pragma unroll
    for (int t = 0; t < MT; ++t) acc[t] = (v8f){};

    const float* bp_base = B + (size_t)khalf * ldb + col_ld;

    for (int kc0 = 0; kc0 < K; kc0 += GEMM_KC) {
        // cooperative load of (16*MT) x 128 A tile (coalesced along K)
        for (int idx = tid; idx < 16 * MT * GEMM_KC; idx += blockDim.x) {
            int r = idx >> 7;
            int c = idx & (GEMM_KC - 1);
            int row = m0 + r;
            if (row >= M) row = M - 1;
            As[r * LDS_STRIDE + c] = A[(size_t)row * lda + kc0 + c];
        }
        __syncthreads();

        const float* bp = bp_base + (size_t)kc0 * ldb;
        #pragma unroll 4
        for (int kk = 0; kk < GEMM_KC; kk += 4) {
            v2f b;
            b.x = bp[0];
            b.y = bp[(size_t)ldb];
            bp += (size_t)4 * ldb;
            #pragma unroll
            for (int t = 0; t < MT; ++t) {
                v2f a;
                const float* ap = &As[(t * 16 + mr) * LDS_STRIDE + kk + khalf];
                a.x = ap[0];
                a.y = ap[1];
                acc[t] = __builtin_amdgcn_wmma_f32_16x16x4_f32(
                    false, a, false, b, (short)0, acc[t], false, false);
            }
        }
        __syncthreads();
    }

    if (!colok) return;
    float bb = bias ? bias[col] : 0.0f;
    #pragma unroll
    for (int t = 0; t < MT; ++t) {
        const int rbase = m0 + t * 16 + ((lane >> 4) << 3);   // +8 for upper half-wave
        #pragma unroll
        for (int r = 0; r < 8; ++r) {
            int row = rbase + r;
            if (row < M) {
                size_t o = (size_t)row * ldc + col;
                float v = acc[t][r] + bb;
                if (accumulate) v += Cout[o];
                Cout[o] = v;
            }
        }
    }
}

// ---------------------------------------------------------------------------
// launcher
// ---------------------------------------------------------------------------
extern "C" void kernel_launch(void* const* d_in, const int* in_sizes, int n_in,
                              void* d_out, int out_size, void* d_ws, size_t ws_size,
                              hipStream_t stream) {
    const float* x     = (const float*)d_in[0];
    const int*   ei    = (const int*)  d_in[1];
    const float* ew    = (const float*)d_in[2];
    const float* W_in  = (const float*)d_in[3];
    const float* b_in  = (const float*)d_in[4];
    const float* W_gcn = (const float*)d_in[5];
    const float* b_gcn = (const float*)d_in[6];
    const float* W_out = (const float*)d_in[7];
    const float* b_out = (const float*)d_in[8];
    float* out = (float*)d_out;

    const int H  = in_sizes[4];              // 128
    const int F  = in_sizes[3] / H;          // 512
    const int Nn = in_sizes[0] / F;          // 50000
    const int E  = in_sizes[1] / 2;          // 800000
    const int C  = in_sizes[8];              // 40
    const int NBASE = 4;

    const int* src = ei;
    const int* dst = ei + E;

    // workspace carve-up (floats)
    float* ws = (float*)d_ws;
    size_t off = 0;
    float* deg   = ws + off; off += (size_t)Nn;        // becomes dinv in-place
    float* normv = ws + off; off += (size_t)E;
    float* h0    = ws + off; off += (size_t)Nn * H;
    float* tA    = ws + off; off += (size_t)Nn * H;
    float* tB    = ws + off; off += (size_t)Nn * H;
    float* hw    = ws + off; off += (size_t)Nn * H;
    (void)ws_size; (void)n_in; (void)out_size;

    // --- gcn_norm ---
    k_fill     <<<cdiv(Nn, 256), 256, 0, stream>>>(deg, 1.0f, Nn);   // self-loop weight
    k_deg_accum<<<cdiv(E, 256),  256, 0, stream>>>(dst, ew, deg, E);
    k_dinv     <<<cdiv(Nn, 256), 256, 0, stream>>>(deg, Nn);
    k_norm     <<<cdiv(E, 256),  256, 0, stream>>>(src, dst, ew, deg, normv, E);

    const int mblocks = cdiv(Nn, 16 * MT);
    const int thrH    = ((H + 15) / 16) * 32;          // 8 waves for H=128
    const int thrC    = 128;                           // 4 waves (3 useful) for C=40

    // --- input linear: h0 = x @ W_in + b_in ---
    k_gemm_f32_wmma<<<mblocks, thrH, 0, stream>>>(x, F, W_in, H, b_in, h0, H,
                                                  Nn, F, H, 0);

    // --- output accumulator: out = b_out + h0 @ W_out[0:H] ---
    k_init_out<<<cdiv((long)Nn * C, 256), 256, 0, stream>>>(out, b_out, Nn, C);
    k_gemm_f32_wmma<<<mblocks, thrC, 0, stream>>>(h0, H, W_out, C, nullptr, out, C,
                                                  Nn, H, C, 1);

    // --- inception branches ---
    int k = 0;
    for (int j = 0; j < NBASE; ++j) {
        const float* cur = h0;
        for (int s = 0; s <= j; ++s) {
            float* nxt = (s & 1) ? tB : tA;
            // hw = cur @ W_gcn[k]
            k_gemm_f32_wmma<<<mblocks, thrH, 0, stream>>>(
                cur, H, W_gcn + (size_t)k * H * H, H, nullptr, hw, H, Nn, H, H, 0);
            // nxt = b_gcn[k] + dinv^2 * hw   (self-loop + bias)
            k_self_init<<<cdiv((long)Nn * (H / 4), 256), 256, 0, stream>>>(
                hw, deg, b_gcn + (size_t)k * H, nxt, Nn, H);
            // nxt[dst] += norm[e] * hw[src]  (one wave per edge, fp32 atomics in L2)
            k_edge_agg<<<cdiv((long)E * 32, 256), 256, 0, stream>>>(
                src, dst, normv, hw, nxt, E, H);
            cur = nxt;
            ++k;
        }
        // out += branch_j @ W_out[(j+1)*H : (j+2)*H]
        k_gemm_f32_wmma<<<mblocks, thrC, 0, stream>>>(
            cur, H, W_out + (size_t)(j + 1) * H * C, C, nullptr, out, C, Nn, H, C, 1);
    }
}